// ComplexMixture_54906861912216
// MI455X (gfx1250) — compile-verified
//
#include <hip/hip_runtime.h>

typedef float v2f __attribute__((ext_vector_type(2)));
typedef float v8f __attribute__((ext_vector_type(8)));

#define NB 16
#define NS 1024
#define ND 768
#define NBC (ND / 128) /* 6 block-cols (and rows) of 128 per batch */

// Each wave32 computes a 32x64 output region (2x4 tiles of 16x16) of both
// out_r and out_i for one batch, using native fp32 WMMA (16x16x4):
//
//   out_r[i,j] = sum_s w[s]*r[s,i]*r[s,j] + w[s]*im[s,i]*im[s,j]
//   out_i[i,j] = sum_s w[s]*im[s,i]*r[s,j] - w[s]*r[s,i]*im[s,j]
//
// Blocks are 8 waves arranged 4x2 -> 128x128 output region, so waves share
// A-side / B-side columns through WGP$/L2 (per-batch input = 6 MB, L2 = 192 MB).
// Per ISA 7.12.2, for 16x16x4 f32 both A (16x4) and B (4x16) fragments have
// per-lane addressing: element {0,1} = mem[(k0 + 2*(lane>>4) + {0,1})*D + col].
__global__ __launch_bounds__(256) void dens_mix_wmma(
    const float* __restrict__ real,
    const float* __restrict__ imag,
    const float* __restrict__ weight,
    float* __restrict__ out) {
  const int lane = threadIdx.x & 31;
  const int wave = threadIdx.x >> 5;
  const int wrow = wave >> 1;                      // 0..3
  const int wcol = wave & 1;                       // 0..1

  const int b  = blockIdx.x / (NBC * NBC);
  const int r2 = blockIdx.x % (NBC * NBC);
  const int i0 = (r2 / NBC) * 128 + wrow * 32;     // output row base (2 I-tiles)
  const int j0 = (r2 % NBC) * 128 + wcol * 64;     // output col base (4 J-tiles)

  const int c  = lane & 15;                        // column within 16-tile
  const int kg = lane >> 4;                        // K half-group (0/1)

  const float* __restrict__ Rb = real   + (long)b * NS * ND;
  const float* __restrict__ Ib = imag   + (long)b * NS * ND;
  const float* __restrict__ Wb = weight + (long)b * NS;

  v8f acc_r[2][4] = {};   // out_r accumulators
  v8f acc_i[2][4] = {};   // out_i accumulators

  for (int k0 = 0; k0 < NS; k0 += 4) {
    const int  ka   = k0 + 2 * kg;                 // this lane's first K row
    const long row0 = (long)ka * ND;
    const long row1 = row0 + ND;
    const float w0 = Wb[ka];
    const float w1 = Wb[ka + 1];

    v2f fwr[2], fwi[2], fnwr[2], fr[4], fim[4];
#pragma unroll
    for (int t = 0; t < 2; ++t) {                  // A-side (rows i), weighted
      const int colI = i0 + t * 16 + c;
      const float ar0 = Rb[row0 + colI];
      const float ar1 = Rb[row1 + colI];
      const float ai0 = Ib[row0 + colI];
      const float ai1 = Ib[row1 + colI];
      fwr[t]  = (v2f){  w0 * ar0,  w1 * ar1 };
      fwi[t]  = (v2f){  w0 * ai0,  w1 * ai1 };
      fnwr[t] = (v2f){ -w0 * ar0, -w1 * ar1 };     // for the out_i minus term
    }
#pragma unroll
    for (int u = 0; u < 4; ++u) {                  // B-side (cols j), raw
      const int colJ = j0 + u * 16 + c;
      fr[u]  = (v2f){ Rb[row0 + colJ], Rb[row1 + colJ] };
      fim[u] = (v2f){ Ib[row0 + colJ], Ib[row1 + colJ] };
    }

#pragma unroll
    for (int t = 0; t < 2; ++t) {
#pragma unroll
      for (int u = 0; u < 4; ++u) {
        // (neg_a, A, neg_b, B, c_mod, C, reuse_a, reuse_b)
        acc_r[t][u] = __builtin_amdgcn_wmma_f32_16x16x4_f32(
            false, fwr[t], false, fr[u], (short)0, acc_r[t][u], false, false);
        acc_r[t][u] = __builtin_amdgcn_wmma_f32_16x16x4_f32(
            false, fwi[t], false, fim[u], (short)0, acc_r[t][u], false, false);
        acc_i[t][u] = __builtin_amdgcn_wmma_f32_16x16x4_f32(
            false, fwi[t], false, fr[u], (short)0, acc_i[t][u], false, false);
        acc_i[t][u] = __builtin_amdgcn_wmma_f32_16x16x4_f32(
            false, fnwr[t], false, fim[u], (short)0, acc_i[t][u], false, false);
      }
    }
  }

  // C/D 16x16 f32 layout: VGPR v, lanes 0-15 -> M=v, lanes 16-31 -> M=v+8; N=lane&15
  float* __restrict__ Or = out + (long)b * ND * ND;
  float* __restrict__ Oi = out + (long)NB * ND * ND + (long)b * ND * ND;
#pragma unroll
  for (int t = 0; t < 2; ++t) {
#pragma unroll
    for (int u = 0; u < 4; ++u) {
      const int col = j0 + u * 16 + c;
#pragma unroll
      for (int v = 0; v < 8; ++v) {
        const int row = i0 + t * 16 + v + 8 * kg;
        Or[(long)row * ND + col] = acc_r[t][u][v];
        Oi[(long)row * ND + col] = acc_i[t][u][v];
      }
    }
  }
}

extern "C" void kernel_launch(void* const* d_in, const int* in_sizes, int n_in,
                              void* d_out, int out_size, void* d_ws, size_t ws_size,
                              hipStream_t stream) {
  (void)in_sizes; (void)n_in; (void)out_size; (void)d_ws; (void)ws_size;
  const float* real   = (const float*)d_in[0];
  const float* imag   = (const float*)d_in[1];
  const float* weight = (const float*)d_in[2];
  float* out = (float*)d_out;

  dim3 grid(NB * NBC * NBC);               // 16 * 36 = 576 blocks
  dim3 block(256);                         // 8 waves, arranged 4x2
  dens_mix_wmma<<<grid, block, 0, stream>>>(real, imag, weight, out);
}